// BertWordPair_70214125355302
// MI455X (gfx1250) — compile-verified
//
#include <hip/hip_runtime.h>
#include <hip/hip_bf16.h>
#include <math.h>

// ---------------------------------------------------------------------------
// Problem constants (from reference): B=4, S=1024, H=768, D=64,
// 8 utterance bands of 128 tokens. token_index = s%128, utterance_index = s/128.
// ---------------------------------------------------------------------------
#define Bsz   4
#define Ssz   1024
#define Hsz   768
#define Dsz   64

typedef float v2f __attribute__((ext_vector_type(2)));
typedef float v8f __attribute__((ext_vector_type(8)));

// D = A(16x4,f32) * B(4x16,f32) + C(16x16,f32)  -- CDNA5 full-precision WMMA
__device__ inline v8f wmma_f32x4(v2f a, v2f b, v8f c) {
    return __builtin_amdgcn_wmma_f32_16x16x4_f32(
        /*neg_a=*/false, a, /*neg_b=*/false, b,
        /*c_mod=*/(short)0, c, /*reuse_a=*/false, /*reuse_b=*/false);
}

// ---------------------------------------------------------------------------
// K1: P = seq(4096x768) @ W(768xN) + bias.  One wave per 16x16 tile.
//   A-frag: lane&15 = M row, lane>>4 selects K pair {2h,2h+1} (contig float2)
//   B-frag: lane&15 = N col, lane>>4 selects K pair (two strided loads)
//   C/D:    reg r -> M = r + 8*(lane>>4), N = lane&15
// ---------------------------------------------------------------------------
__global__ void proj_gemm(const float* __restrict__ A, const float* __restrict__ W,
                          const float* __restrict__ bias, float* __restrict__ P,
                          int N, int tiles_n) {
    const int lane = threadIdx.x & 31;
    const int wave = threadIdx.x >> 5;
    const int tile = blockIdx.x * (blockDim.x >> 5) + wave;
    const int tm = tile / tiles_n;
    const int tn = tile - tm * tiles_n;
    const int m0 = tm << 4, n0 = tn << 4;
    const int l = lane & 15, hi = lane >> 4;

    const float* __restrict__ arow = A + (size_t)(m0 + l) * Hsz;
    v8f acc = {};
    for (int k = 0; k < Hsz; k += 4) {
        v2f af, bf;
        af.x = arow[k + 2 * hi];
        af.y = arow[k + 2 * hi + 1];
        bf.x = W[(size_t)(k + 2 * hi) * N + n0 + l];
        bf.y = W[(size_t)(k + 2 * hi + 1) * N + n0 + l];
        acc = wmma_f32x4(af, bf, acc);
    }
    const float bb = bias[n0 + l];
#pragma unroll
    for (int r = 0; r < 8; ++r) {
        const int m = m0 + r + 8 * hi;
        P[(size_t)m * N + n0 + l] = acc[r] + bb;
    }
}

// ---------------------------------------------------------------------------
// K2: RoPE rotation.  One wave per (b,c,s) row; lane i handles d-pair (2i,2i+1).
// Produces qp/qm/kp/km for token angle (base 10000) and, if use_utt, for
// utterance angle (base 15).  Output layout: [(b*C+c)*S + s]*64 + d.
// ---------------------------------------------------------------------------
__global__ void rope_rotate(const float* __restrict__ P, int C, int N,
                            float* __restrict__ Qtp, float* __restrict__ Qtm,
                            float* __restrict__ Ktp, float* __restrict__ Ktm,
                            float* __restrict__ Qup, float* __restrict__ Qum,
                            float* __restrict__ Kup, float* __restrict__ Kum,
                            int use_utt) {
    const int i   = threadIdx.x & 31;  // d-pair index 0..31
    const int row = (blockIdx.x * blockDim.x + threadIdx.x) >> 5;
    const int b   = row / (C * Ssz);
    const int rem = row - b * (C * Ssz);
    const int c   = rem >> 10;
    const int s   = rem & (Ssz - 1);

    const float* __restrict__ src = P + (size_t)(b * Ssz + s) * N + c * (4 * Dsz);
    const float qtx = src[2 * i],            qty = src[2 * i + 1];
    const float qux = src[Dsz + 2 * i],      quy = src[Dsz + 2 * i + 1];
    const float ktx = src[2 * Dsz + 2 * i],  kty = src[2 * Dsz + 2 * i + 1];
    const float kux = src[3 * Dsz + 2 * i],  kuy = src[3 * Dsz + 2 * i + 1];

    const size_t dst = ((size_t)(b * C + c) * Ssz + s) * Dsz + 2 * i;

    // token angle: inv = 10000^(-i/32)
    {
        const float inv = __expf(-(float)i * (9.2103403720f / 32.0f));
        const float ang = (float)(s & 127) * inv;
        const float sn = __sinf(ang), cs = __cosf(ang);
        Qtp[dst] = qtx * cs - qty * sn;  Qtp[dst + 1] = qty * cs + qtx * sn;
        Qtm[dst] = qtx * cs + qty * sn;  Qtm[dst + 1] = qty * cs - qtx * sn;
        Ktp[dst] = ktx * cs - kty * sn;  Ktp[dst + 1] = kty * cs + ktx * sn;
        Ktm[dst] = ktx * cs + kty * sn;  Ktm[dst + 1] = kty * cs - ktx * sn;
    }
    // utterance angle: inv = 15^(-i/32)
    if (use_utt) {
        const float inv = __expf(-(float)i * (2.7080502011f / 32.0f));
        const float ang = (float)(s >> 7) * inv;
        const float sn = __sinf(ang), cs = __cosf(ang);
        Qup[dst] = qux * cs - quy * sn;  Qup[dst + 1] = quy * cs + qux * sn;
        Qum[dst] = qux * cs + quy * sn;  Qum[dst + 1] = quy * cs - qux * sn;
        Kup[dst] = kux * cs - kuy * sn;  Kup[dst + 1] = kuy * cs + kux * sn;
        Kum[dst] = kux * cs + kuy * sn;  Kum[dst + 1] = kuy * cs - kux * sn;
    }
}

// ---------------------------------------------------------------------------
// K3: pairwise logits + fused log-softmax/NLL loss.
// One wave per 16x16 output tile.  Tiles never straddle the 128-row bands,
// so the flip-select is uniform per tile:
//   tid_m>0 && tid_m<tid_n  -> qm . kp   (L_mp)
//   tid_m==0 && tid_n>0     -> qp . km   (L_pm)
//   else                    -> qp . kp   (L_pp)
// ---------------------------------------------------------------------------
template <int C, bool UTT>
__global__ void pair_logits_loss(
    const float* __restrict__ Qtp, const float* __restrict__ Qtm,
    const float* __restrict__ Ktp, const float* __restrict__ Ktm,
    const float* __restrict__ Qup, const float* __restrict__ Qum,
    const float* __restrict__ Kup, const float* __restrict__ Kum,
    const int* __restrict__ labels, float* __restrict__ tags,
    float* __restrict__ loss) {
    const int lane = threadIdx.x & 31;
    const int wave = threadIdx.x >> 5;
    const int tile = blockIdx.x * (blockDim.x >> 5) + wave;
    const int b   = tile >> 12;          // 64*64 = 4096 tiles per batch
    const int rem = tile & 4095;
    const int tm = rem >> 6, tn = rem & 63;
    const int m0 = tm << 4, n0 = tn << 4;
    const int tidm = m0 >> 7, tidn = n0 >> 7;

    const float *Qt, *Kt, *Qu = nullptr, *Ku = nullptr;
    if (tidm > 0 && tidm < tidn) {
        Qt = Qtm; Kt = Ktp; if (UTT) { Qu = Qum; Ku = Kup; }
    } else if (tidm == 0 && tidn > 0) {
        Qt = Qtp; Kt = Ktm; if (UTT) { Qu = Qup; Ku = Kum; }
    } else {
        Qt = Qtp; Kt = Ktp; if (UTT) { Qu = Qup; Ku = Kup; }
    }

    const int l = lane & 15, hi = lane >> 4;
    v8f acc[C];
#pragma unroll
    for (int c = 0; c < C; ++c) {
        v8f a = {};
        const size_t qr = ((size_t)(b * C + c) * Ssz + m0 + l) * Dsz;
        const size_t kr = ((size_t)(b * C + c) * Ssz + n0 + l) * Dsz;
        const float* __restrict__ q = Qt + qr;
        const float* __restrict__ k = Kt + kr;
#pragma unroll
        for (int kk = 0; kk < Dsz; kk += 4) {
            v2f af, bf;
            af.x = q[kk + 2 * hi]; af.y = q[kk + 2 * hi + 1];
            bf.x = k[kk + 2 * hi]; bf.y = k[kk + 2 * hi + 1];
            a = wmma_f32x4(af, bf, a);
        }
        if (UTT) {
            const float* __restrict__ qu = Qu + qr;
            const float* __restrict__ ku = Ku + kr;
#pragma unroll
            for (int kk = 0; kk < Dsz; kk += 4) {
                v2f af, bf;
                af.x = qu[kk + 2 * hi]; af.y = qu[kk + 2 * hi + 1];
                bf.x = ku[kk + 2 * hi]; bf.y = ku[kk + 2 * hi + 1];
                a = wmma_f32x4(af, bf, a);
            }
        }
        acc[c] = a;
    }

    // Store logits (C innermost) + fused NLL (weights/masks are all ones).
    float nll_sum = 0.0f;
#pragma unroll
    for (int r = 0; r < 8; ++r) {
        const int m = m0 + r + 8 * hi;
        const int n = n0 + l;
        float lv[C];
        float mx = -3.0e38f;
#pragma unroll
        for (int c = 0; c < C; ++c) { lv[c] = acc[c][r]; mx = fmaxf(mx, lv[c]); }
        float se = 0.0f;
#pragma unroll
        for (int c = 0; c < C; ++c) se += __expf(lv[c] - mx);
        const float lse = mx + __logf(se);

        const size_t eidx = ((size_t)(b * Ssz + m)) * Ssz + n;
        const int lab = labels[eidx];
        float lsel = lv[0];
#pragma unroll
        for (int c = 1; c < C; ++c) lsel = (lab == c) ? lv[c] : lsel;
        nll_sum += lse - lsel;

        float* __restrict__ o = tags + eidx * C;
#pragma unroll
        for (int c = 0; c < C; ++c) o[c] = lv[c];
    }
    // wave32 reduction then one atomic per wave
    for (int off = 16; off > 0; off >>= 1) nll_sum += __shfl_down(nll_sum, off, 32);
    if (lane == 0)
        atomicAdd(loss, nll_sum * (1.0f / ((float)Bsz * Ssz * Ssz)));
}

__global__ void zero_losses(float* __restrict__ p) {
    if (threadIdx.x < 3) p[threadIdx.x] = 0.0f;
}

// ---------------------------------------------------------------------------
// Host-side orchestration
// ---------------------------------------------------------------------------
extern "C" void kernel_launch(void* const* d_in, const int* in_sizes, int n_in,
                              void* d_out, int out_size, void* d_ws, size_t ws_size,
                              hipStream_t stream) {
    const float* seq   = (const float*)d_in[0];
    const float* W_ent = (const float*)d_in[1];
    const float* b_ent = (const float*)d_in[2];
    const float* W_rel = (const float*)d_in[3];
    const float* b_rel = (const float*)d_in[4];
    const float* W_pol = (const float*)d_in[5];
    const float* b_pol = (const float*)d_in[6];
    const int* ent_m   = (const int*)d_in[9];
    const int* rel_m   = (const int*)d_in[10];
    const int* pol_m   = (const int*)d_in[11];

    float* out = (float*)d_out;
    float* ws  = (float*)d_ws;

    // ---- workspace layout (floats) ----
    const size_t P_ent_off = 0;
    const size_t P_rel_off = P_ent_off + (size_t)Bsz * Ssz * 1024;   //  4.19M
    const size_t P_pol_off = P_rel_off + (size_t)Bsz * Ssz * 768;    //  3.15M
    size_t cur = P_pol_off + (size_t)Bsz * Ssz * 1024;

    const size_t rotE = (size_t)Bsz * 4 * Ssz * Dsz;  // per ent rot array
    const size_t rotR = (size_t)Bsz * 3 * Ssz * Dsz;
    const size_t rotP = rotE;
    float* eQtp = ws + cur; cur += rotE; float* eQtm = ws + cur; cur += rotE;
    float* eKtp = ws + cur; cur += rotE; float* eKtm = ws + cur; cur += rotE;
    float* rQtp = ws + cur; cur += rotR; float* rQtm = ws + cur; cur += rotR;
    float* rKtp = ws + cur; cur += rotR; float* rKtm = ws + cur; cur += rotR;
    float* rQup = ws + cur; cur += rotR; float* rQum = ws + cur; cur += rotR;
    float* rKup = ws + cur; cur += rotR; float* rKum = ws + cur; cur += rotR;
    float* pQtp = ws + cur; cur += rotP; float* pQtm = ws + cur; cur += rotP;
    float* pKtp = ws + cur; cur += rotP; float* pKtm = ws + cur; cur += rotP;
    float* pQup = ws + cur; cur += rotP; float* pQum = ws + cur; cur += rotP;
    float* pKup = ws + cur; cur += rotP; float* pKum = ws + cur; cur += rotP;

    float* P_ent = ws + P_ent_off;
    float* P_rel = ws + P_rel_off;
    float* P_pol = ws + P_pol_off;

    // ---- output layout (floats): [3 losses][tags0][tags1][tags2] ----
    float* tags0 = out + 3;
    float* tags1 = tags0 + (size_t)Bsz * Ssz * Ssz * 4;
    float* tags2 = tags1 + (size_t)Bsz * Ssz * Ssz * 3;

    zero_losses<<<1, 32, 0, stream>>>(out);

    // K1: projections (256 row-tiles of 16; 4 waves per block)
    proj_gemm<<<(256 * 64) / 4, 128, 0, stream>>>(seq, W_ent, b_ent, P_ent, 1024, 64);
    proj_gemm<<<(256 * 48) / 4, 128, 0, stream>>>(seq, W_rel, b_rel, P_rel, 768, 48);
    proj_gemm<<<(256 * 64) / 4, 128, 0, stream>>>(seq, W_pol, b_pol, P_pol, 1024, 64);

    // K2: RoPE rotation (rows = B*C*S; 32 lanes per row, 256 threads/block)
    rope_rotate<<<(Bsz * 4 * Ssz * 32) / 256, 256, 0, stream>>>(
        P_ent, 4, 1024, eQtp, eQtm, eKtp, eKtm,
        nullptr, nullptr, nullptr, nullptr, 0);
    rope_rotate<<<(Bsz * 3 * Ssz * 32) / 256, 256, 0, stream>>>(
        P_rel, 3, 768, rQtp, rQtm, rKtp, rKtm, rQup, rQum, rKup, rKum, 1);
    rope_rotate<<<(Bsz * 4 * Ssz * 32) / 256, 256, 0, stream>>>(
        P_pol, 4, 1024, pQtp, pQtm, pKtp, pKtm, pQup, pQum, pKup, pKum, 1);

    // K3: pairwise logits + fused loss (B*64*64 = 16384 tiles; 4 waves/block)
    pair_logits_loss<4, false><<<16384 / 4, 128, 0, stream>>>(
        eQtp, eQtm, eKtp, eKtm, nullptr, nullptr, nullptr, nullptr,
        ent_m, tags0, out + 0);
    pair_logits_loss<3, true><<<16384 / 4, 128, 0, stream>>>(
        rQtp, rQtm, rKtp, rKtm, rQup, rQum, rKup, rKum,
        rel_m, tags1, out + 1);
    pair_logits_loss<4, true><<<16384 / 4, 128, 0, stream>>>(
        pQtp, pQtm, pKtp, pKtm, pQup, pQum, pKup, pKum,
        pol_m, tags2, out + 2);
}